// Layer_22574348108153
// MI455X (gfx1250) — compile-verified
//
#include <hip/hip_runtime.h>

// Problem constants (match reference)
#define B_       128
#define T_       2048
#define D_       256
#define TILE     32
#define NTILES   (T_ / TILE)
#define NTHREADS 256
#define BD       (B_ * D_)

static_assert(D_ == NTHREADS, "one thread per feature dim");

typedef float v2f __attribute__((ext_vector_type(2)));
typedef float v8f __attribute__((ext_vector_type(8)));
// Exact pointee type the async builtin wants: 16-byte int vector
typedef int v4i_vs __attribute__((vector_size(16)));
typedef __attribute__((address_space(1))) v4i_vs* gas1_v4i;
typedef __attribute__((address_space(3))) v4i_vs* las3_v4i;

// ---------------- CDNA5 feature guards (compile-safe) ----------------
#if __has_builtin(__builtin_amdgcn_global_load_async_to_lds_b128)
#define USE_ASYNC 1
#else
#define USE_ASYNC 0
#endif

#if __has_builtin(__builtin_amdgcn_s_wait_asynccnt)
#define WAIT_ASYNC(n) __builtin_amdgcn_s_wait_asynccnt(n)
#elif defined(__AMDGCN__)
#define WAIT_ASYNC(n) asm volatile("s_wait_asynccnt %0" ::"i"(n) : "memory")
#else
#define WAIT_ASYNC(n)
#endif

// Stage one 32-row x 256-col f32 tile (32KB) global -> LDS.
// 8 waves x 8 instrs x (32 lanes x 16B) = 32KB, fully coalesced 512B bursts.
// Async path: 8 outstanding ASYNCcnt ops per wave per tile.
__device__ __forceinline__ void issue_tile_load(const float* __restrict__ gsrc,
                                                float* lds_dst, int tid) {
  const int wave = tid >> 5, lane = tid & 31;
#pragma unroll
  for (int j = 0; j < 8; ++j) {
    const int fidx = wave * 1024 + j * 128 + lane * 4;  // float index
#if USE_ASYNC
    __builtin_amdgcn_global_load_async_to_lds_b128(
        (gas1_v4i)(gsrc + fidx), (las3_v4i)(lds_dst + fidx),
        /*offset=*/0, /*cpol=*/0);
#else
    *(float4*)(lds_dst + fidx) = *(const float4*)(gsrc + fidx);
#endif
  }
}

__device__ __forceinline__ float wave_sum(float v) {
#pragma unroll
  for (int off = 16; off; off >>= 1) v += __shfl_xor(v, off, 32);
  return v;
}

// ---------------- Kernel 1: mids = aspect @ W_mul^T via V_WMMA_F32_16X16X4_F32
// C[m=batch, n=out_dim] = sum_k aspect[m,k] * W_mul[n,k].  M=128,N=256,K=256.
// 128 16x16 tiles -> 16 blocks x 8 waves, 64 WMMA steps (K=4 each) per tile.
__global__ __launch_bounds__(256) void mids_gemm(const float* __restrict__ aspect,
                                                 const float* __restrict__ Wmul,
                                                 float* __restrict__ mids) {
#if __has_builtin(__builtin_amdgcn_wmma_f32_16x16x4_f32)
  const int wave = threadIdx.x >> 5;
  const int lane = threadIdx.x & 31;
  const int tile = blockIdx.x * 8 + wave;  // 0..127
  const int mi = tile & 7;                 // 8 tiles along M=128
  const int ni = tile >> 3;                // 16 tiles along N=256
  const int row = mi * 16 + (lane & 15);   // A row (batch)
  const int col = ni * 16 + (lane & 15);   // B col (out dim)
  const int khalf = (lane >> 4) * 2;       // lanes 16-31 hold K+2,K+3
  v8f c = {};
#pragma unroll 4
  for (int k0 = 0; k0 < D_; k0 += 4) {
    const int kb = k0 + khalf;
    v2f a, b;
    a.x = aspect[row * D_ + kb];
    a.y = aspect[row * D_ + kb + 1];
    b.x = Wmul[col * D_ + kb];   // B[k,n] = W_mul[n,k]
    b.y = Wmul[col * D_ + kb + 1];
    c = __builtin_amdgcn_wmma_f32_16x16x4_f32(false, a, false, b, (short)0, c,
                                              false, false);
  }
  // D layout: VGPR r -> M = r (lanes 0-15) / r+8 (lanes 16-31), N = lane%16
  const int rbase = mi * 16 + ((lane >> 4) ? 8 : 0);
  const int ccol = ni * 16 + (lane & 15);
#pragma unroll
  for (int r = 0; r < 8; ++r) mids[(rbase + r) * D_ + ccol] = c[r];
#else
  // Scalar fallback (also what the host pass parses). Same launch shape.
  const int gtid = blockIdx.x * 256 + threadIdx.x;
  for (int o = gtid; o < BD; o += 16 * 256) {
    const int b = o >> 8, i = o & 255;
    float s = 0.f;
    for (int j = 0; j < D_; ++j) s += aspect[b * D_ + j] * Wmul[i * D_ + j];
    mids[o] = s;
  }
#endif
}

// ---------------- Kernel 2: fused attention, one block per batch ----------------
// Phase A: stream aspect_memory[b] once:  e_t = exp(tanh(row_t . mids + b_mul)),
//          accA[d] += e_t*row_t[d], sumA += e_t;  new_aspect = accA/sumA.
// Phase B: stream sentiment_memory[b] once: e1 = exp(row . w_ss[:256])*mask,
//          e2 = exp(row . w_sa[:256])*mask (query terms cancel exactly),
//          two weighted accumulators; out = 0.5*acc1/s1 + 0.5*acc2/s2.
__global__ __launch_bounds__(256) void fused_attn(
    const float* __restrict__ aspect, const float* __restrict__ smem_g,
    const float* __restrict__ amem_g, const float* __restrict__ mask,
    const float* __restrict__ mids_g, const float* __restrict__ b_mul,
    const float* __restrict__ w_ss, const float* __restrict__ w_sa,
    float* __restrict__ out) {
  __shared__ __align__(16) float tileA[TILE * D_];  // 32KB
  __shared__ __align__(16) float tileB[TILE * D_];  // 32KB
  __shared__ float mask_s[T_];                      // 8KB
  __shared__ float evals[TILE];
  __shared__ float e1s[TILE];
  __shared__ float e2s[TILE];
  __shared__ float red[8], red2[8];

  const int b = blockIdx.x;
  const int tid = threadIdx.x;
  const int wave = tid >> 5, lane = tid & 31;
  const float* am = amem_g + (size_t)b * T_ * D_;
  const float* sm = smem_g + (size_t)b * T_ * D_;

  // per-lane slices of the three 256-length weight vectors (lane owns dims 8l..8l+7)
  float mid_r[8], wss_r[8], wsa_r[8];
#pragma unroll
  for (int i = 0; i < 8; ++i) {
    mid_r[i] = mids_g[b * D_ + lane * 8 + i];
    wss_r[i] = w_ss[lane * 8 + i];
    wsa_r[i] = w_sa[lane * 8 + i];
  }
  const float bmul = b_mul[0];
  for (int i = tid; i < T_; i += NTHREADS) mask_s[i] = mask[(size_t)b * T_ + i];

  // =========================== Phase A: aspect path ===========================
  float accA = 0.f, sumA = 0.f;
  issue_tile_load(am, tileA, tid);
  issue_tile_load(am + TILE * D_, tileB, tid);
  for (int it = 0; it < NTILES; ++it) {
    float* buf = (it & 1) ? tileB : tileA;
    if (it + 1 < NTILES) { WAIT_ASYNC(8); } else { WAIT_ASYNC(0); }
    __syncthreads();
// scores: 4 rows per wave; lane l covers cols 8l..8l+7
#pragma unroll
    for (int rr = 0; rr < 4; ++rr) {
      const int r = wave * 4 + rr;
      const float4* rp = (const float4*)(buf + r * D_ + lane * 8);
      const float4 x0 = rp[0], x1 = rp[1];
      float s = x0.x * mid_r[0] + x0.y * mid_r[1] + x0.z * mid_r[2] +
                x0.w * mid_r[3] + x1.x * mid_r[4] + x1.y * mid_r[5] +
                x1.z * mid_r[6] + x1.w * mid_r[7];
      s = wave_sum(s);
      const float e = __expf(tanhf(s + bmul));  // bounded -> no max-sub needed
      if (lane == 0) evals[r] = e;
      sumA += e;  // identical in all lanes of this wave
    }
    __syncthreads();
// weighted accumulation: thread tid owns feature dim tid (conflict-free)
#pragma unroll 8
    for (int tt = 0; tt < TILE; ++tt) accA += evals[tt] * buf[tt * D_ + tid];
    __syncthreads();
    if (it + 2 < NTILES) issue_tile_load(am + (size_t)(it + 2) * TILE * D_, buf, tid);
  }
  if (lane == 0) red[wave] = sumA;  // per-wave partials (lane-uniform)
  __syncthreads();
  const float totA = red[0] + red[1] + red[2] + red[3] + red[4] + red[5] +
                     red[6] + red[7];
  const float new_aspect_d = accA / totA;
  out[BD + b * D_ + tid] = aspect[b * D_ + tid] + new_aspect_d;  // aspect_out
  __syncthreads();

  // ========================= Phase B: sentiment path =========================
  float acc1 = 0.f, acc2 = 0.f, sum1 = 0.f, sum2 = 0.f;
  issue_tile_load(sm, tileA, tid);
  issue_tile_load(sm + TILE * D_, tileB, tid);
  for (int it = 0; it < NTILES; ++it) {
    float* buf = (it & 1) ? tileB : tileA;
    if (it + 1 < NTILES) { WAIT_ASYNC(8); } else { WAIT_ASYNC(0); }
    __syncthreads();
#pragma unroll
    for (int rr = 0; rr < 4; ++rr) {
      const int r = wave * 4 + rr;
      const int t = it * TILE + r;
      const float4* rp = (const float4*)(buf + r * D_ + lane * 8);
      const float4 x0 = rp[0], x1 = rp[1];
      float sA = x0.x * wss_r[0] + x0.y * wss_r[1] + x0.z * wss_r[2] +
                 x0.w * wss_r[3] + x1.x * wss_r[4] + x1.y * wss_r[5] +
                 x1.z * wss_r[6] + x1.w * wss_r[7];
      float sB = x0.x * wsa_r[0] + x0.y * wsa_r[1] + x0.z * wsa_r[2] +
                 x0.w * wsa_r[3] + x1.x * wsa_r[4] + x1.y * wsa_r[5] +
                 x1.z * wsa_r[6] + x1.w * wsa_r[7];
      sA = wave_sum(sA);
      sB = wave_sum(sB);
      const float mv = mask_s[t];
      const float e1 = __expf(sA) * mv;  // query const cancels exactly
      const float e2 = __expf(sB) * mv;
      if (lane == 0) { e1s[r] = e1; e2s[r] = e2; }
      sum1 += e1;
      sum2 += e2;
    }
    __syncthreads();
#pragma unroll 8
    for (int tt = 0; tt < TILE; ++tt) {
      const float v = buf[tt * D_ + tid];
      acc1 += e1s[tt] * v;
      acc2 += e2s[tt] * v;
    }
    __syncthreads();
    if (it + 2 < NTILES) issue_tile_load(sm + (size_t)(it + 2) * TILE * D_, buf, tid);
  }
  if (lane == 0) { red[wave] = sum1; red2[wave] = sum2; }
  __syncthreads();
  const float tot1 = red[0] + red[1] + red[2] + red[3] + red[4] + red[5] +
                     red[6] + red[7];
  const float tot2 = red2[0] + red2[1] + red2[2] + red2[3] + red2[4] + red2[5] +
                     red2[6] + red2[7];
  // sentiment_out[b,0,:]  (lambda = 0.5)
  out[b * D_ + tid] = 0.5f * (acc1 / tot1) + 0.5f * (acc2 / tot2);
}

extern "C" void kernel_launch(void* const* d_in, const int* in_sizes, int n_in,
                              void* d_out, int out_size, void* d_ws, size_t ws_size,
                              hipStream_t stream) {
  (void)in_sizes; (void)n_in; (void)out_size; (void)ws_size;
  // setup_inputs order:
  // 0 sentiment (unused: its attention contribution cancels), 1 aspect,
  // 2 sentiment_memory, 3 aspect_memory, 4 mask, 5 W_mul, 6 b_mul,
  // 7 w_ss, 8 b_ss (cancels), 9 w_sa, 10 b_sa (cancels)
  const float* aspect = (const float*)d_in[1];
  const float* smem = (const float*)d_in[2];
  const float* amem = (const float*)d_in[3];
  const float* mask = (const float*)d_in[4];
  const float* Wmul = (const float*)d_in[5];
  const float* bmul = (const float*)d_in[6];
  const float* wss = (const float*)d_in[7];
  const float* wsa = (const float*)d_in[9];
  float* out = (float*)d_out;
  float* mids = (float*)d_ws;  // B*D floats = 128KB scratch

  mids_gemm<<<dim3(16), dim3(256), 0, stream>>>(aspect, Wmul, mids);
  fused_attn<<<dim3(B_), dim3(256), 0, stream>>>(aspect, smem, amem, mask, mids,
                                                 bmul, wss, wsa, out);
}